// CompareModels_81527069213346
// MI455X (gfx1250) — compile-verified
//
#include <hip/hip_runtime.h>
#include <hip/hip_bf16.h>
#include <math.h>

#define BB   8
#define TT   1024
#define DDIM 512
#define HH   8
#define DHH  64
#define DFFN 2048
#define MM   (BB * TT)      /* 8192 rows */
#define EPSF 1e-5f

typedef float v2f __attribute__((ext_vector_type(2)));
typedef float v8f __attribute__((ext_vector_type(8)));

__device__ __forceinline__ v8f wmma_f32(v2f a, v2f b, v8f c) {
  // D = A(16x4 f32) * B(4x16 f32) + C(16x16 f32), exact fp32
  return __builtin_amdgcn_wmma_f32_16x16x4_f32(false, a, false, b, (short)0, c,
                                               false, false);
}

// Async global->LDS copy of 16 bytes per lane (VFLAT GLOBAL_LOAD_ASYNC_TO_LDS_B128,
// tracked by ASYNCcnt). ldsaddr is the 32-bit LDS byte address (low 32 bits of
// the generic pointer), gaddr the per-lane global address.
__device__ __forceinline__ void async_copy_b128(uint32_t ldsaddr,
                                                const float* gaddr) {
  asm volatile("global_load_async_to_lds_b128 %0, %1, off"
               :
               : "v"(ldsaddr), "v"(gaddr)
               : "memory");
}
__device__ __forceinline__ void wait_asynccnt_le4() {
  asm volatile("s_wait_asynccnt 0x4" ::: "memory");
}
__device__ __forceinline__ void wait_asynccnt_0() {
  asm volatile("s_wait_asynccnt 0x0" ::: "memory");
}

// ---------------------------------------------------------------------------
// Generic GEMM: Y[M,N] = act(X[M,K] @ W[N,K]^T + bias) (* Emul)
// ACT: 0=none, 1=gelu(exact), 2=sigmoid. Block tile 64x64, 8 waves.
// Double-buffered LDS staging via async global->LDS copies.
// ---------------------------------------------------------------------------
template <int ACT>
__global__ __launch_bounds__(256) void gemm_kernel(
    const float* __restrict__ X, const float* __restrict__ W,
    const float* __restrict__ bias, const float* __restrict__ Emul,
    float* __restrict__ Y, int N, int K) {
  __shared__ __align__(16) float As[2][64 * 36];
  __shared__ __align__(16) float Bs[2][64 * 36];
  const int tid  = threadIdx.x;
  const int wave = tid >> 5;
  const int lane = tid & 31;
  const int hl   = lane >> 4;   // K-pair half
  const int lm   = lane & 15;   // row/col within fragment
  const int wm   = wave & 3;    // 4 m-subtiles of 16
  const int wn   = wave >> 2;   // 2 n-halves of 32
  const int bM   = blockIdx.y * 64;
  const int bN   = blockIdx.x * 64;

  // this thread's two float4 slots of the 64x32 stage tile
  const int row0 = tid >> 3;             // 0..31
  const int c40  = (tid & 7) << 2;       // 0..28
  const int row1 = row0 + 32;            // 32..63
  const int c41  = c40;

  v8f acc0 = {};
  v8f acc1 = {};

  const int nstage = K >> 5;  // K/32

  // prologue: stage 0
  {
    async_copy_b128((uint32_t)(uintptr_t)&As[0][row0 * 36 + c40],
                    &X[(size_t)(bM + row0) * K + c40]);
    async_copy_b128((uint32_t)(uintptr_t)&As[0][row1 * 36 + c41],
                    &X[(size_t)(bM + row1) * K + c41]);
    async_copy_b128((uint32_t)(uintptr_t)&Bs[0][row0 * 36 + c40],
                    &W[(size_t)(bN + row0) * K + c40]);
    async_copy_b128((uint32_t)(uintptr_t)&Bs[0][row1 * 36 + c41],
                    &W[(size_t)(bN + row1) * K + c41]);
  }

  for (int s = 0; s < nstage; ++s) {
    const int buf = s & 1;
    if (s + 1 < nstage) {
      const int nb = buf ^ 1;
      const int kc = (s + 1) << 5;
      async_copy_b128((uint32_t)(uintptr_t)&As[nb][row0 * 36 + c40],
                      &X[(size_t)(bM + row0) * K + kc + c40]);
      async_copy_b128((uint32_t)(uintptr_t)&As[nb][row1 * 36 + c41],
                      &X[(size_t)(bM + row1) * K + kc + c41]);
      async_copy_b128((uint32_t)(uintptr_t)&Bs[nb][row0 * 36 + c40],
                      &W[(size_t)(bN + row0) * K + kc + c40]);
      async_copy_b128((uint32_t)(uintptr_t)&Bs[nb][row1 * 36 + c41],
                      &W[(size_t)(bN + row1) * K + kc + c41]);
      wait_asynccnt_le4();   // current stage's 4 copies have completed
    } else {
      wait_asynccnt_0();
    }
    __syncthreads();

    const float* Ab = As[buf];
    const float* Bb = Bs[buf];
#pragma unroll
    for (int kk = 0; kk < 32; kk += 4) {
      int k  = kk + 2 * hl;
      v2f a  = *(const v2f*)&Ab[(wm * 16 + lm) * 36 + k];
      v2f b0 = *(const v2f*)&Bb[(wn * 32 + lm) * 36 + k];
      v2f b1 = *(const v2f*)&Bb[(wn * 32 + 16 + lm) * 36 + k];
      acc0 = wmma_f32(a, b0, acc0);
      acc1 = wmma_f32(a, b1, acc1);
    }
    __syncthreads();
  }

#pragma unroll
  for (int r = 0; r < 8; ++r) {
    int m = bM + wm * 16 + r + 8 * hl;
#pragma unroll
    for (int j = 0; j < 2; ++j) {
      int n   = bN + wn * 32 + j * 16 + lm;
      float v = (j ? acc1[r] : acc0[r]) + bias[n];
      if (ACT == 1) v = 0.5f * v * (1.0f + erff(v * 0.70710678118654752f));
      else if (ACT == 2) v = 1.0f / (1.0f + expf(-v));
      if (Emul) v *= Emul[(size_t)m * N + n];
      Y[(size_t)m * N + n] = v;
    }
  }
}

// ---------------------------------------------------------------------------
// Flash attention: one wave per (b, h, 16-row q tile). softmax(QK^T/8) V.
// qkv layout [B,T,3D]: q at +0, k at +D, v at +2D; head h occupies cols h*64.
// ---------------------------------------------------------------------------
__global__ __launch_bounds__(256) void attn_kernel(
    const float* __restrict__ qkv, float* __restrict__ ctx) {
  __shared__ float Pl[8][16][17];
  const int tid  = threadIdx.x;
  const int wave = tid >> 5;
  const int lane = tid & 31;
  const int hl   = lane >> 4;
  const int lm   = lane & 15;
  const int task = blockIdx.x * 8 + wave;  // 0..4095
  const int qt   = task & 63;
  const int bh   = task >> 6;
  const int h    = bh & 7;
  const int b    = bh >> 3;

  const float* Qrow =
      qkv + ((size_t)(b * TT + qt * 16 + lm)) * (3 * DDIM) + h * DHH;
  v2f qa[16];
#pragma unroll
  for (int kc = 0; kc < 16; ++kc) {
    int k = kc * 4 + 2 * hl;
    v2f q;
    q.x    = Qrow[k];
    q.y    = Qrow[k + 1];
    qa[kc] = q * 0.125f;  // 1/sqrt(64) folded into Q
  }

  float rm[8], rl[8];
  v8f o0 = {}, o1 = {}, o2 = {}, o3 = {};
#pragma unroll
  for (int r = 0; r < 8; ++r) {
    rm[r] = -INFINITY;
    rl[r] = 0.0f;
  }

  for (int kt = 0; kt < 64; ++kt) {
    const float* Krow =
        qkv + ((size_t)(b * TT + kt * 16 + lm)) * (3 * DDIM) + DDIM + h * DHH;
    v8f s = {};
#pragma unroll
    for (int kc = 0; kc < 16; ++kc) {
      int k = kc * 4 + 2 * hl;
      v2f kb;
      kb.x = Krow[k];
      kb.y = Krow[k + 1];
      s    = wmma_f32(qa[kc], kb, s);
    }
    // online softmax: row m = r + 8*hl lives in one 16-lane group
#pragma unroll
    for (int r = 0; r < 8; ++r) {
      float v = s[r];
#pragma unroll
      for (int off = 1; off < 16; off <<= 1)
        v = fmaxf(v, __shfl_xor(v, off, 16));
      float nm = fmaxf(rm[r], v);
      float p  = expf(s[r] - nm);
      float ps = p;
#pragma unroll
      for (int off = 1; off < 16; off <<= 1) ps += __shfl_xor(ps, off, 16);
      float sc = expf(rm[r] - nm);
      rl[r]    = rl[r] * sc + ps;
      rm[r]    = nm;
      o0[r] *= sc;
      o1[r] *= sc;
      o2[r] *= sc;
      o3[r] *= sc;
      s[r] = p;
    }
    // transpose P (C-layout -> A-layout) through per-wave LDS
#pragma unroll
    for (int r = 0; r < 8; ++r) Pl[wave][r + 8 * hl][lm] = s[r];
    asm volatile("" ::: "memory");
    v2f pa[4];
#pragma unroll
    for (int c = 0; c < 4; ++c) {
      int k   = c * 4 + 2 * hl;
      pa[c].x = Pl[wave][lm][k];
      pa[c].y = Pl[wave][lm][k + 1];
    }
    asm volatile("" ::: "memory");
    const float* Vb =
        qkv + ((size_t)(b * TT + kt * 16)) * (3 * DDIM) + 2 * DDIM + h * DHH;
#pragma unroll
    for (int c = 0; c < 4; ++c) {
      int key          = c * 4 + 2 * hl;
      const float* V0  = Vb + (size_t)key * (3 * DDIM);
      const float* V1  = V0 + 3 * DDIM;
      v2f vb;
      vb.x = V0[lm];      vb.y = V1[lm];      o0 = wmma_f32(pa[c], vb, o0);
      vb.x = V0[16 + lm]; vb.y = V1[16 + lm]; o1 = wmma_f32(pa[c], vb, o1);
      vb.x = V0[32 + lm]; vb.y = V1[32 + lm]; o2 = wmma_f32(pa[c], vb, o2);
      vb.x = V0[48 + lm]; vb.y = V1[48 + lm]; o3 = wmma_f32(pa[c], vb, o3);
    }
  }

#pragma unroll
  for (int r = 0; r < 8; ++r) {
    int m       = qt * 16 + r + 8 * hl;
    float inv   = 1.0f / rl[r];
    float* Crow = ctx + ((size_t)(b * TT + m)) * DDIM + h * DHH;
    Crow[lm]      = o0[r] * inv;
    Crow[16 + lm] = o1[r] * inv;
    Crow[32 + lm] = o2[r] * inv;
    Crow[48 + lm] = o3[r] * inv;
  }
}

// ---------------------------------------------------------------------------
// Fused (residual|multiply) + LayerNorm over rows of 512. MODE 0: a+r, 1: a*r
// ---------------------------------------------------------------------------
template <int MODE>
__global__ __launch_bounds__(256) void ln_kernel(
    const float* __restrict__ a, const float* __restrict__ r,
    const float* __restrict__ g, const float* __restrict__ be,
    float* __restrict__ y) {
  __shared__ float red[8];
  __shared__ float red2[8];
  const int tid = threadIdx.x;
  size_t base   = (size_t)blockIdx.x * DDIM;
  float v0 = a[base + tid], v1 = a[base + tid + 256];
  float r0 = r[base + tid], r1 = r[base + tid + 256];
  if (MODE == 0) { v0 += r0; v1 += r1; } else { v0 *= r0; v1 *= r1; }
  float s = v0 + v1;
#pragma unroll
  for (int off = 16; off; off >>= 1) s += __shfl_xor(s, off, 32);
  if ((tid & 31) == 0) red[tid >> 5] = s;
  __syncthreads();
  if (tid == 0) {
    float t = 0.0f;
    for (int i = 0; i < 8; ++i) t += red[i];
    red[0] = t * (1.0f / DDIM);
  }
  __syncthreads();
  float mu = red[0];
  float d0 = v0 - mu, d1 = v1 - mu;
  float q = d0 * d0 + d1 * d1;
#pragma unroll
  for (int off = 16; off; off >>= 1) q += __shfl_xor(q, off, 32);
  if ((tid & 31) == 0) red2[tid >> 5] = q;
  __syncthreads();
  if (tid == 0) {
    float t = 0.0f;
    for (int i = 0; i < 8; ++i) t += red2[i];
    red2[0] = t * (1.0f / DDIM);
  }
  __syncthreads();
  float rs = rsqrtf(red2[0] + EPSF);
  y[base + tid]       = d0 * rs * g[tid] + be[tid];
  y[base + tid + 256] = d1 * rs * g[tid + 256] + be[tid + 256];
}

// ---------------------------------------------------------------------------
// Mamba scan: h_t = tanh(A h_{t-1} + u_t), one block per batch, h in LDS.
// ---------------------------------------------------------------------------
__global__ __launch_bounds__(256) void scan_kernel(
    const float* __restrict__ A, const float* __restrict__ u,
    float* __restrict__ hs) {
  __shared__ __align__(16) float hsh[DDIM];
  const int b   = blockIdx.x;
  const int tid = threadIdx.x;
  const int d0  = tid * 2;
  hsh[d0]     = 0.0f;
  hsh[d0 + 1] = 0.0f;
  __syncthreads();
  const float* A0 = A + (size_t)d0 * DDIM;
  const float* A1 = A0 + DDIM;
  for (int t = 0; t < TT; ++t) {
    float acc0 = 0.0f, acc1 = 0.0f;
#pragma unroll 4
    for (int k = 0; k < DDIM; k += 4) {
      float4 hv = *(const float4*)&hsh[k];
      float4 a0 = *(const float4*)&A0[k];
      float4 a1 = *(const float4*)&A1[k];
      acc0 += a0.x * hv.x + a0.y * hv.y + a0.z * hv.z + a0.w * hv.w;
      acc1 += a1.x * hv.x + a1.y * hv.y + a1.z * hv.z + a1.w * hv.w;
    }
    size_t ro = ((size_t)b * TT + t) * DDIM;
    float h0  = tanhf(acc0 + u[ro + d0]);
    float h1  = tanhf(acc1 + u[ro + d0 + 1]);
    __syncthreads();
    hsh[d0]     = h0;
    hsh[d0 + 1] = h1;
    hs[ro + d0]     = h0;
    hs[ro + d0 + 1] = h1;
    __syncthreads();
  }
}

// ---------------------------------------------------------------------------
extern "C" void kernel_launch(void* const* d_in, const int* in_sizes, int n_in,
                              void* d_out, int out_size, void* d_ws,
                              size_t ws_size, hipStream_t stream) {
  const float* x    = (const float*)d_in[0];
  const float* Wqkv = (const float*)d_in[1];
  const float* bqkv = (const float*)d_in[2];
  const float* Wo   = (const float*)d_in[3];
  const float* bo   = (const float*)d_in[4];
  const float* ln1g = (const float*)d_in[5];
  const float* ln1b = (const float*)d_in[6];
  const float* W1   = (const float*)d_in[7];
  const float* b1   = (const float*)d_in[8];
  const float* W2   = (const float*)d_in[9];
  const float* b2   = (const float*)d_in[10];
  const float* ln2g = (const float*)d_in[11];
  const float* ln2b = (const float*)d_in[12];
  const float* Amat = (const float*)d_in[13];
  const float* WB   = (const float*)d_in[14];
  const float* bB   = (const float*)d_in[15];
  const float* WC   = (const float*)d_in[16];
  const float* bC   = (const float*)d_in[17];
  const float* Wg   = (const float*)d_in[18];
  const float* bg   = (const float*)d_in[19];
  const float* lnmg = (const float*)d_in[20];
  const float* lnmb = (const float*)d_in[21];

  float* out_trans = (float*)d_out;
  float* out_mamba = out_trans + (size_t)MM * DDIM;

  float* ws   = (float*)d_ws;
  size_t off  = 0;
  float* qkv  = ws + off; off += (size_t)MM * 3 * DDIM;
  float* ctx  = ws + off; off += (size_t)MM * DDIM;
  float* x1   = ws + off; off += (size_t)MM * DDIM;
  float* hbuf = ws + off; off += (size_t)MM * DFFN;
  float* tmp  = ws + off; off += (size_t)MM * DDIM;
  float* Bt   = ws + off; off += (size_t)MM * DDIM;
  float* Ct   = ws + off; off += (size_t)MM * DDIM;
  float* ubuf = ws + off; off += (size_t)MM * DDIM;
  float* hsb  = ws + off;

  dim3 blk(256);

  // ---- transformer branch ----
  gemm_kernel<0><<<dim3(3 * DDIM / 64, MM / 64), blk, 0, stream>>>(
      x, Wqkv, bqkv, nullptr, qkv, 3 * DDIM, DDIM);
  attn_kernel<<<dim3(512), blk, 0, stream>>>(qkv, ctx);
  gemm_kernel<0><<<dim3(DDIM / 64, MM / 64), blk, 0, stream>>>(
      ctx, Wo, bo, nullptr, tmp, DDIM, DDIM);
  ln_kernel<0><<<dim3(MM), blk, 0, stream>>>(x, tmp, ln1g, ln1b, x1);
  gemm_kernel<1><<<dim3(DFFN / 64, MM / 64), blk, 0, stream>>>(
      x1, W1, b1, nullptr, hbuf, DFFN, DDIM);
  gemm_kernel<0><<<dim3(DDIM / 64, MM / 64), blk, 0, stream>>>(
      hbuf, W2, b2, nullptr, tmp, DDIM, DFFN);
  ln_kernel<0><<<dim3(MM), blk, 0, stream>>>(x1, tmp, ln2g, ln2b, out_trans);

  // ---- mamba branch ----
  gemm_kernel<0><<<dim3(DDIM / 64, MM / 64), blk, 0, stream>>>(
      x, WB, bB, nullptr, Bt, DDIM, DDIM);
  gemm_kernel<0><<<dim3(DDIM / 64, MM / 64), blk, 0, stream>>>(
      x, WC, bC, nullptr, Ct, DDIM, DDIM);
  gemm_kernel<2><<<dim3(DDIM / 64, MM / 64), blk, 0, stream>>>(
      x, Wg, bg, Bt, ubuf, DDIM, DDIM);  // u = sigmoid(xWg+bg) * Bt
  scan_kernel<<<dim3(BB), blk, 0, stream>>>(Amat, ubuf, hsb);
  ln_kernel<1><<<dim3(MM), blk, 0, stream>>>(Ct, hsb, lnmg, lnmb, out_mamba);
}